// EnsembleClustering_62646392979777
// MI455X (gfx1250) — compile-verified
//
#include <hip/hip_runtime.h>
#include <math.h>

typedef __attribute__((ext_vector_type(2))) float v2f;
typedef __attribute__((ext_vector_type(8))) float v8f;

#define EPS 1e-6f
#define RSQRT48 0.14433756729740643f

// ---------------------------------------------------------------------------
// Kernel 1: pool x (16,384,56,56) -> xp (384 x 848), column n = b*53 + p
//   p in [0,49): 7x7 adaptive pool (8x8 blocks);  p in [49,53): 2x2 pool (28x28)
// One block (64 threads) per (b,c) plane.
// ---------------------------------------------------------------------------
__global__ void ec_pool_kernel(const float* __restrict__ x, float* __restrict__ xp) {
    const int plane = blockIdx.x;          // b*384 + c
    const int b = plane / 384;
    const int c = plane % 384;
    const float* __restrict__ px = x + (size_t)plane * 3136;
    const int tid = threadIdx.x;

    // 7x7 pool: 49 tiles of 8x8
    if (tid < 49) {
        const int th = tid / 7, tw = tid % 7;
        float s = 0.f;
        #pragma unroll
        for (int r = 0; r < 8; ++r) {
            const float4* rp = (const float4*)(px + (th * 8 + r) * 56 + tw * 8);
            float4 a = rp[0], bb = rp[1];
            s += a.x + a.y + a.z + a.w + bb.x + bb.y + bb.z + bb.w;
        }
        xp[c * 848 + b * 53 + tid] = s * (1.0f / 64.0f);
    }

    // 2x2 pool: 4 quadrants of 28x28, 16 threads each
    const int q  = tid >> 4;      // quadrant 0..3
    const int r  = tid & 15;
    const int qi = q >> 1, qj = q & 1;
    float s2 = 0.f;
    for (int row = r; row < 28; row += 16) {
        const float4* rp = (const float4*)(px + (qi * 28 + row) * 56 + qj * 28);
        #pragma unroll
        for (int j = 0; j < 7; ++j) {
            float4 v = rp[j];
            s2 += v.x + v.y + v.z + v.w;
        }
    }
    #pragma unroll
    for (int off = 8; off >= 1; off >>= 1)
        s2 += __shfl_xor(s2, off, 16);
    if (r == 0)
        xp[c * 848 + b * 53 + 49 + q] = s2 * (1.0f / 784.0f);
}

// ---------------------------------------------------------------------------
// Kernel 2/4: fp32 WMMA GEMM:  C[M,N] = A[M,K] * B[K,N] + bias[M]
// Wave-per-16x16-tile, V_WMMA_F32_16X16X4_F32, K-loop step 4.
// A fragment (ISA 7.12.2): lane L holds row M=L%16, VGPR r holds K=r+2*(L/16)
// C/D fragment:            lane L holds col N=L%16, VGPR r holds M=r+8*(L/16)
// ---------------------------------------------------------------------------
__global__ void ec_gemm_wmma_kernel(const float* __restrict__ A,
                                    const float* __restrict__ B,
                                    const float* __restrict__ bias,
                                    float* __restrict__ C,
                                    int M, int K, int N) {
    const int lane = threadIdx.x & 31;
    const int wave = threadIdx.x >> 5;
    const int n0 = blockIdx.x * 16;
    const int m0 = (blockIdx.y * 8 + wave) * 16;
    if (m0 >= M || n0 >= N) return;     // wave-uniform: EXEC stays all-ones

    const int half = lane >> 4;         // 0/1
    const int l16  = lane & 15;

    v8f acc = {};
    const float* __restrict__ Arow = A + (size_t)(m0 + l16) * K + 2 * half;
    const float* __restrict__ Bcol = B + n0 + l16 + (size_t)(2 * half) * N;

    #pragma unroll 4
    for (int k0 = 0; k0 < K; k0 += 4) {
        v2f a = *(const v2f*)(Arow + k0);        // A(m, k0+2h), A(m, k0+2h+1)
        v2f bfrag;
        bfrag.x = Bcol[(size_t)k0 * N];          // B(k0+2h,   n)
        bfrag.y = Bcol[(size_t)k0 * N + N];      // B(k0+2h+1, n)
        acc = __builtin_amdgcn_wmma_f32_16x16x4_f32(
            false, a, false, bfrag, (short)0, acc, false, false);
    }

    #pragma unroll
    for (int r = 0; r < 8; ++r) {
        const int m = m0 + r + 8 * half;
        C[(size_t)m * N + n0 + l16] = acc[r] + bias[m];
    }
}

// ---------------------------------------------------------------------------
// Kernel 3: per (batch*head) clustering attention.
// y: (1920 x 848) rows o = g*384 + head*48 + c, cols n = b16*53 + p
//    p<49: 7x7 agents, p=49+m: 2x2 clusters.
// out7: (384 x 784) rows d = head*48 + c, cols n = b16*49 + p
// One block (64 threads) per bh in [0,128).
// ---------------------------------------------------------------------------
__global__ void ec_attn_kernel(const float* __restrict__ y,
                               const float* __restrict__ sim_alpha,
                               const float* __restrict__ sim_beta,
                               float* __restrict__ out7) {
    __shared__ float s_kc[2][4][48];
    __shared__ float s_vc[2][4][48];
    __shared__ float s_scores[4][49];
    __shared__ float s_prob[4][49];
    __shared__ float s_agg[8][48];
    __shared__ float s_assign[8][49];
    __shared__ float s_rowsum[4];
    __shared__ float s_rnorm[8];
    __shared__ float s_ab[16];

    const int bh   = blockIdx.x;
    const int b16  = bh >> 3;
    const int head = bh & 7;
    const int tid  = threadIdx.x;
    const size_t base_n = (size_t)b16 * 53;
    const int hc = head * 48;

    if (tid < 16)
        s_ab[tid] = (tid < 8) ? sim_alpha[tid] : sim_beta[tid - 8];

    // load kc/vc for both modules (2 kinds * 2 modules * 4 m * 48 c = 768)
    for (int idx = tid; idx < 768; idx += 64) {
        const int kind = idx >> 9 ? 1 : (idx >= 384);  // 0 = kc, 1 = vc
        const int t = idx & 383;
        const int i = t / 192;
        const int m = (t / 48) & 3;
        const int c = t % 48;
        const int g = (kind ? (2 + 2 * i) : (1 + 2 * i));
        const float v = y[((size_t)(g * 384 + hc + c)) * 848 + base_n + 49 + m];
        if (kind) s_vc[i][m][c] = v; else s_kc[i][m][c] = v;
    }
    __syncthreads();

    for (int i = 0; i < 2; ++i) {
        const int gk = 1 + 2 * i, gv = 2 + 2 * i;
        // scores[m][p]
        if (tid < 49) {
            float s[4] = {0.f, 0.f, 0.f, 0.f};
            for (int c = 0; c < 48; ++c) {
                const float kv = y[((size_t)(gk * 384 + hc + c)) * 848 + base_n + tid];
                #pragma unroll
                for (int m = 0; m < 4; ++m) s[m] += s_kc[i][m][c] * kv;
            }
            #pragma unroll
            for (int m = 0; m < 4; ++m) s_scores[m][tid] = s[m] * RSQRT48;
        }
        __syncthreads();

        if (i == 0) {
            // softmax over p (axis -1), per m
            if (tid < 4) {
                float mx = -1e30f;
                for (int p = 0; p < 49; ++p) mx = fmaxf(mx, s_scores[tid][p]);
                float sum = 0.f;
                for (int p = 0; p < 49; ++p) {
                    const float e = __expf(s_scores[tid][p] - mx);
                    s_prob[tid][p] = e; sum += e;
                }
                const float inv = 1.0f / sum;
                for (int p = 0; p < 49; ++p) s_prob[tid][p] *= inv;
            }
        } else {
            // softmax over m (axis 1), per p
            if (tid < 49) {
                float mx = -1e30f;
                #pragma unroll
                for (int m = 0; m < 4; ++m) mx = fmaxf(mx, s_scores[m][tid]);
                float e[4], sum = 0.f;
                #pragma unroll
                for (int m = 0; m < 4; ++m) { e[m] = __expf(s_scores[m][tid] - mx); sum += e[m]; }
                const float inv = 1.0f / sum;
                #pragma unroll
                for (int m = 0; m < 4; ++m) s_prob[m][tid] = e[m] * inv;
            }
        }
        __syncthreads();
        if (i == 1 && tid < 4) {
            float rs = 0.f;
            for (int p = 0; p < 49; ++p) rs += s_prob[tid][p];
            s_rowsum[tid] = rs;
        }
        __syncthreads();

        // agg[m][c] (+vc), 192 (m,c) pairs over 64 threads
        for (int pair = tid; pair < 192; pair += 64) {
            const int m = pair / 48, c = pair % 48;
            const float* vrow = y + ((size_t)(gv * 384 + hc + c)) * 848 + base_n;
            float s = 0.f;
            for (int p = 0; p < 49; ++p) s += s_prob[m][p] * vrow[p];
            if (i == 1) s /= (s_rowsum[m] + EPS);
            s_agg[i * 4 + m][c] = s + s_vc[i][m][c];
        }
        __syncthreads();
    }

    // row norms of agg
    if (tid < 8) {
        float s = 0.f;
        for (int c = 0; c < 48; ++c) s += s_agg[tid][c] * s_agg[tid][c];
        s_rnorm[tid] = 1.0f / (sqrtf(s) + EPS);
    }
    __syncthreads();

    // sim / assignment, per agent pixel p
    if (tid < 49) {
        float t[8] = {0.f, 0.f, 0.f, 0.f, 0.f, 0.f, 0.f, 0.f};
        float pn = 0.f;
        for (int c = 0; c < 48; ++c) {
            const float pa = y[((size_t)(hc + c)) * 848 + base_n + tid];   // group 0
            pn += pa * pa;
            #pragma unroll
            for (int m = 0; m < 8; ++m) t[m] += s_agg[m][c] * pa;
        }
        const float rp = 1.0f / (sqrtf(pn) + EPS);
        float simv[8], mx = -1e30f;
        #pragma unroll
        for (int m = 0; m < 8; ++m) {
            simv[m] = s_ab[m] * (t[m] * s_rnorm[m] * rp) + s_ab[8 + m];
            mx = fmaxf(mx, simv[m]);
        }
        float sum = 0.f;
        #pragma unroll
        for (int m = 0; m < 8; ++m) { simv[m] = __expf(simv[m] - mx); sum += simv[m]; }
        const float inv = 1.0f / sum;
        #pragma unroll
        for (int m = 0; m < 8; ++m) s_assign[m][tid] = simv[m] * inv;
    }
    __syncthreads();

    // out7[head*48+c][b16*49+p] = sum_m agg[m][c]*assign[m][p]
    if (tid < 49) {
        float asg[8];
        #pragma unroll
        for (int m = 0; m < 8; ++m) asg[m] = s_assign[m][tid];
        for (int c = 0; c < 48; ++c) {
            float o = 0.f;
            #pragma unroll
            for (int m = 0; m < 8; ++m) o += s_agg[m][c] * asg[m];
            out7[(size_t)(hc + c) * 784 + (size_t)b16 * 49 + tid] = o;
        }
    }
}

// ---------------------------------------------------------------------------
// Kernel 5: bilinear resize 7x7 -> 56x56 (half-pixel, edge clamp == jax linear)
// out2: (384 x 784), cols n = b*49 + pix.  out: (16,384,56,56)
// ---------------------------------------------------------------------------
__global__ void ec_resize_kernel(const float* __restrict__ out2,
                                 float* __restrict__ out) {
    __shared__ float src[49];
    const int plane = blockIdx.x;          // b*384 + o
    const int b = plane / 384;
    const int o = plane % 384;
    const int tid = threadIdx.x;
    if (tid < 49) src[tid] = out2[(size_t)o * 784 + (size_t)b * 49 + tid];
    __syncthreads();

    float* __restrict__ dst = out + (size_t)plane * 3136;
    for (int idx = tid; idx < 3136; idx += 256) {
        const int h = idx / 56, w = idx % 56;
        const float fh = (h + 0.5f) * 0.125f - 0.5f;
        const float fw = (w + 0.5f) * 0.125f - 0.5f;
        const float h0f = floorf(fh), w0f = floorf(fw);
        const float ah = fh - h0f, aw = fw - w0f;
        const int h0 = min(6, max(0, (int)h0f));
        const int h1 = min(6, max(0, (int)h0f + 1));
        const int w0 = min(6, max(0, (int)w0f));
        const int w1 = min(6, max(0, (int)w0f + 1));
        const float v = (1.f - ah) * ((1.f - aw) * src[h0 * 7 + w0] + aw * src[h0 * 7 + w1])
                      +        ah  * ((1.f - aw) * src[h1 * 7 + w0] + aw * src[h1 * 7 + w1]);
        dst[idx] = v;
    }
}

// ---------------------------------------------------------------------------
static inline size_t ec_align256(size_t v) { return (v + 255) & ~(size_t)255; }

extern "C" void kernel_launch(void* const* d_in, const int* in_sizes, int n_in,
                              void* d_out, int out_size, void* d_ws, size_t ws_size,
                              hipStream_t stream) {
    const float* x         = (const float*)d_in[0];
    const float* proj_w    = (const float*)d_in[1];
    const float* proj_b    = (const float*)d_in[2];
    const float* sim_alpha = (const float*)d_in[3];
    const float* sim_beta  = (const float*)d_in[4];
    const float* proj2_w   = (const float*)d_in[5];
    const float* proj2_b   = (const float*)d_in[6];
    float* out = (float*)d_out;

    char* ws = (char*)d_ws;
    size_t off = 0;
    float* xp   = (float*)(ws + off); off += ec_align256((size_t)384  * 848 * 4);
    float* y    = (float*)(ws + off); off += ec_align256((size_t)1920 * 848 * 4);
    float* out7 = (float*)(ws + off); off += ec_align256((size_t)384  * 784 * 4);
    float* out2 = (float*)(ws + off); off += ec_align256((size_t)384  * 784 * 4);
    (void)ws_size; (void)in_sizes; (void)n_in; (void)out_size;

    // 1) pool x -> xp (384 x 848)
    ec_pool_kernel<<<dim3(16 * 384), dim3(64), 0, stream>>>(x, xp);

    // 2) y = proj_w(1920x384) * xp(384x848) + proj_b   (WMMA f32)
    ec_gemm_wmma_kernel<<<dim3(848 / 16, 120 / 8), dim3(256), 0, stream>>>(
        proj_w, xp, proj_b, y, 1920, 384, 848);

    // 3) attention per (batch*head) -> out7 (384 x 784)
    ec_attn_kernel<<<dim3(128), dim3(64), 0, stream>>>(y, sim_alpha, sim_beta, out7);

    // 4) out2 = proj2_w(384x384) * out7(384x784) + proj2_b   (WMMA f32)
    ec_gemm_wmma_kernel<<<dim3(784 / 16, 24 / 8), dim3(256), 0, stream>>>(
        proj2_w, out7, proj2_b, out2, 384, 384, 784);

    // 5) bilinear resize to (16,384,56,56)
    ec_resize_kernel<<<dim3(16 * 384), dim3(256), 0, stream>>>(out2, out);
}